// Aggregation_47648367182711
// MI455X (gfx1250) — compile-verified
//
#include <hip/hip_runtime.h>

// Problem constants (fixed by the reference).
#define KSZ   5
#define PADSZ 2
#define B_    16
#define C_    512
#define CW    64
#define G_    8
#define HW    64            // H = W = Hout = Wout
#define NPIX  4096          // 64*64
#define TROWS 16            // output rows per block
#define LROWS 20            // TROWS + 2*PAD halo rows
#define LW    72            // LDS row stride (4 left pad + 64 + 4 right pad) -> 288B rows, 16B aligned
#define LDSBUF (LROWS * LW) // 1440 floats per buffer
#define DUMMY_OFF (2 * LDSBUF) // dummy 4-float slot for masked-off async writes

typedef float v4f __attribute__((ext_vector_type(4)));
typedef int   v4i __attribute__((vector_size(16)));
typedef v4i __attribute__((address_space(1)))* as1_v4i_ptr;   // global B128 pointer
typedef v4i __attribute__((address_space(3)))* as3_v4i_ptr;   // LDS    B128 pointer

#if defined(__has_builtin)
# if __has_builtin(__builtin_amdgcn_global_load_async_to_lds_b128)
#  define HAVE_ASYNC_LDS 1
# endif
# if __has_builtin(__builtin_amdgcn_s_wait_asynccnt)
#  define HAVE_WAIT_ASYNC 1
# endif
#endif

template <int N>
__device__ __forceinline__ void wait_asynccnt_le() {
#if defined(HAVE_ASYNC_LDS)
# if defined(HAVE_WAIT_ASYNC)
    __builtin_amdgcn_s_wait_asynccnt(N);
# else
    asm volatile("s_wait_asynccnt %0" ::"i"(N) : "memory");
# endif
#endif
}

// Stage the (clipped) 20x64 input tile of one channel plane into LDS buffer
// lds[bufb .. bufb+1439], interior at columns 4..67. Halo stays pre-zeroed.
// Every thread issues exactly 2 async B128 ops (uniform per-wave ASYNCcnt):
// invalid lanes read a clamped (valid) global address and land in a dummy slot.
__device__ __forceinline__ void stage_group(const float* __restrict__ gin,
                                            float* lds, int ho0, int bufb, int tid) {
#if defined(HAVE_ASYNC_LDS)
#pragma unroll
    for (int k = 0; k < 2; ++k) {
        int e_raw = k * 256 + tid;                       // 0..511; 320 real float4 cells
        int e     = (e_raw < 320) ? e_raw : (e_raw - 320);
        int lr    = e >> 4;                              // LDS row 0..19
        int c4    = (e & 15) << 2;                       // interior col 0,4,...,60
        int hh    = ho0 - PADSZ + lr;                    // global input row
        bool ok   = (e_raw < 320) && (hh >= 0) && (hh < HW);
        int hcl   = hh < 0 ? 0 : (hh > HW - 1 ? HW - 1 : hh);
        const float* src = gin + hcl * HW + c4;          // always a valid address
        int dofs  = ok ? (bufb + lr * LW + 4 + c4) : DUMMY_OFF;
        __builtin_amdgcn_global_load_async_to_lds_b128(
            (as1_v4i_ptr)(src), (as3_v4i_ptr)(lds + dofs), 0, 0);
    }
#else
    // Fallback: plain global load + LDS store (DScnt-ordered; __syncthreads covers it).
#pragma unroll
    for (int k = 0; k < 2; ++k) {
        int e_raw = k * 256 + tid;
        if (e_raw < 320) {
            int lr = e_raw >> 4;
            int c4 = (e_raw & 15) << 2;
            int hh = ho0 - PADSZ + lr;
            if (hh >= 0 && hh < HW) {
                v4f v = *(const v4f*)(gin + hh * HW + c4);
                *(v4f*)(lds + bufb + lr * LW + 4 + c4) = v;
            }
        }
    }
#endif
}

__global__ __launch_bounds__(256) void agg_kernel(const float* __restrict__ in,
                                                  const float* __restrict__ wgt,
                                                  float* __restrict__ out) {
    __shared__ __align__(16) float lds[2 * LDSBUF + 4];

    const int tid = threadIdx.x;
    const unsigned bid = blockIdx.x;       // 4096 blocks = b(16) * cw(64) * rowtile(4)
    const int rt = bid & 3;
    const int cw = (bid >> 2) & 63;
    const int b  = bid >> 8;

    const int row = tid >> 4;              // 0..15 : output row within tile
    const int wo0 = (tid & 15) << 2;       // 0,4,...,60 : first of 4 output cols
    const int ho0 = rt * TROWS;
    const int ho  = ho0 + row;
    const int p0  = ho * HW + wo0;         // flat pixel index of this thread's quad

    // 25 per-pixel weight quads: streamed once (NT), reused across all 8 groups.
    const float* wbase = wgt + (size_t)((b * CW + cw) * (KSZ * KSZ)) * NPIX + p0;
    v4f wv[KSZ * KSZ];
#pragma unroll
    for (int idx = 0; idx < KSZ * KSZ; ++idx)
        wv[idx] = __builtin_nontemporal_load((const v4f*)(wbase + idx * NPIX));

    // Zero LDS once: halo/clipped rows/dummy stay zero; interior is rewritten per group.
    for (int i = tid; i < 2 * LDSBUF + 4; i += 256) lds[i] = 0.0f;
    __syncthreads();   // zero-stores committed before any async write lands

    const float* inb = in + (size_t)(b * C_ + cw) * NPIX;  // group-0 plane for this cw

    // Prologue: stage group 0 into buffer 0.
    stage_group(inb, lds, ho0, 0, tid);

    for (int g = 0; g < G_; ++g) {
        // Kick the DMA for the next group into the other buffer (double-buffered).
        if (g + 1 < G_)
            stage_group(inb + (size_t)(g + 1) * CW * NPIX, lds, ho0,
                        ((g + 1) & 1) * LDSBUF, tid);

        // Retire this group's async copies (per-wave, counts are uniform), then
        // barrier so every wave's tile data is visible.
        if (g + 1 < G_) wait_asynccnt_le<2>();
        else            wait_asynccnt_le<0>();
        __syncthreads();

        const float* xb = lds + (g & 1) * LDSBUF;
        v4f acc = {0.0f, 0.0f, 0.0f, 0.0f};
#pragma unroll
        for (int i = 0; i < KSZ; ++i) {
            const float* rp = xb + (row + i) * LW + wo0;   // 16B-aligned
            v4f q0 = *(const v4f*)(rp);
            v4f q1 = *(const v4f*)(rp + 4);
            v4f q2 = *(const v4f*)(rp + 8);
            float xr[12];
            *(v4f*)(&xr[0]) = q0;
            *(v4f*)(&xr[4]) = q1;
            *(v4f*)(&xr[8]) = q2;
#pragma unroll
            for (int j = 0; j < KSZ; ++j) {
                v4f w = wv[i * KSZ + j];
                acc.x = __builtin_fmaf(xr[2 + j + 0], w.x, acc.x);
                acc.y = __builtin_fmaf(xr[2 + j + 1], w.y, acc.y);
                acc.z = __builtin_fmaf(xr[2 + j + 2], w.z, acc.z);
                acc.w = __builtin_fmaf(xr[2 + j + 3], w.w, acc.w);
            }
        }

        float* op = out + (size_t)(b * C_ + g * CW + cw) * NPIX + p0;
        __builtin_nontemporal_store(acc, (v4f*)op);

        // All waves done reading this buffer before iteration g+1 stages group g+2 into it.
        __syncthreads();
    }
}

extern "C" void kernel_launch(void* const* d_in, const int* in_sizes, int n_in,
                              void* d_out, int out_size, void* d_ws, size_t ws_size,
                              hipStream_t stream) {
    (void)in_sizes; (void)n_in; (void)out_size; (void)d_ws; (void)ws_size;
    const float* inp = (const float*)d_in[0];
    const float* wgt = (const float*)d_in[1];
    float* out = (float*)d_out;

    dim3 grid(B_ * CW * (HW / TROWS));  // 16*64*4 = 4096 blocks
    dim3 block(256);                    // 8 wave32 waves / block
    hipLaunchKernelGGL(agg_kernel, grid, block, 0, stream, inp, wgt, out);
}